// RegionLoss_1829656068458
// MI455X (gfx1250) — compile-verified
//
#include <hip/hip_runtime.h>
#include <hip/hip_bf16.h>
#include <stdint.h>

// ---------------------------------------------------------------------------
// RegionLoss for MI455X (gfx1250).
// x: (B, NA*14, GH, GW) f32,  targets: (B, T, 5) f32  ->  scalar f32 loss.
// Bandwidth-bound (46MB main tensor @ 23.3TB/s ~ 2.5us floor), so:
//   * stream each contiguous 28KB (b,a) tile into LDS via
//     global_load_async_to_lds_b128 (CDNA5 async path, ASYNCcnt)
//   * compact target scatter (index grid) instead of 7 full f32 grids
//   * block reduction epilogue via chained V_WMMA_F32_16X16X4_F32 with a
//     ones A-matrix: D = 1(16x4) x B(4x16) + C  == exact f32 column sums.
// ---------------------------------------------------------------------------

#define NA 5
#define NC 7
#define GH 16
#define GW 32
#define CELLS_PER_TILE (GH * GW)          // 512
#define CH 14
#define TILE_BYTES (CH * CELLS_PER_TILE * 4)  // 28672
#define NQ 9                               // # of partial-sum quantities

typedef float v2f __attribute__((ext_vector_type(2)));
typedef float v8f __attribute__((ext_vector_type(8)));

__device__ __constant__ float c_AW[NA] = {1.0f, 2.0f, 3.5f, 5.0f, 8.0f};
__device__ __constant__ float c_AH[NA] = {1.5f, 3.0f, 4.5f, 6.0f, 10.0f};

__device__ __forceinline__ float sigm(float x) {
    return 1.0f / (1.0f + __expf(-x));
}
__device__ __forceinline__ float logc(float p) {
    return __logf(fmaxf(p, 1e-12f));
}

// ---------------------------------------------------------------- K1: init --
__global__ void rl_init(int* __restrict__ tIdxG, int* __restrict__ ignG,
                        float* __restrict__ accW, int ncells) {
    int i = blockIdx.x * blockDim.x + threadIdx.x;
    if (i < ncells) {
        tIdxG[i] = -1;
        ignG[i]  = 0;
    }
    if (i < 16) accW[i] = 0.0f;
}

// ------------------------------------------------------ K2: target scatter --
__global__ void rl_build_targets(const float* __restrict__ tgt,
                                 int* __restrict__ tIdxG,
                                 int* __restrict__ ignG,
                                 int B, int T) {
    int t = blockIdx.x * blockDim.x + threadIdx.x;
    if (t >= B * T) return;
    int b  = t / T;
    int ti = t - b * T;
    const float* tg = tgt + (size_t)t * 5;
    float s5 = tg[0] + tg[1] + tg[2] + tg[3] + tg[4];
    if (!(s5 > 0.0f)) return;                      // invalid target
    float gx = tg[1] * (float)GW;
    float gy = tg[2] * (float)GH;
    float gw = tg[3] * (float)GW;
    float gh = tg[4] * (float)GH;
    int gi = (int)floorf(gx);
    int gj = (int)floorf(gy);
    gi = min(max(gi, 0), GW - 1);
    gj = min(max(gj, 0), GH - 1);
    float best = -1.0f;
    int   ba   = 0;
#pragma unroll
    for (int a = 0; a < NA; ++a) {
        float aw = c_AW[a], ah = c_AH[a];
        float inter = fminf(gw, aw) * fminf(gh, ah);
        float iou   = inter / (gw * gh + aw * ah - inter + 1e-16f);
        if (iou > best) { best = iou; ba = a; }
        if (iou > 0.6f) {                          // ignore-mask scatter
            ignG[(((b * NA + a) * GH) + gj) * GW + gi] = 1;
        }
    }
    tIdxG[(((b * NA + ba) * GH) + gj) * GW + gi] = ti;  // positive cell
}

// ------------------------------------------- K3: fused loss partial sums ----
__global__ __launch_bounds__(512)
void rl_loss(const float* __restrict__ x, const float* __restrict__ tgt,
             const int* __restrict__ tIdxG, const int* __restrict__ ignG,
             float* __restrict__ accW, int T) {
    __shared__ __align__(16) float sx[CH * CELLS_PER_TILE];  // 28KB tile
    __shared__ float part[16 * 16];                          // 16 waves x 16 q

    const int tid  = threadIdx.x;
    const int tile = blockIdx.x;            // tile = b*NA + a
    const int a    = tile % NA;
    const int b    = tile / NA;

    // ---- async stream the contiguous 28KB tile into LDS (ASYNCcnt path) ----
    {
        const char* gbase = (const char*)x + (size_t)tile * TILE_BYTES;
        unsigned lbase = (unsigned)(unsigned long long)(uintptr_t)sx;
#pragma unroll
        for (int r = 0; r < 4; ++r) {
            int idx = r * 512 + tid;        // 16B granules, 1792 total
            if (idx < TILE_BYTES / 16) {
                unsigned    loff = lbase + (unsigned)idx * 16u;
                const char* gsrc = gbase + (size_t)idx * 16;
                asm volatile("global_load_async_to_lds_b128 %0, %1, off"
                             :: "v"(loff), "v"(gsrc) : "memory");
            }
        }
        asm volatile("s_wait_asynccnt 0" ::: "memory");
    }
    if (tid < 256) part[tid] = 0.0f;
    __syncthreads();

    // ---- per-cell loss math (wave32 VALU + trans ops) ----
    float p[CH];
#pragma unroll
    for (int c = 0; c < CH; ++c) p[c] = sx[c * CELLS_PER_TILE + tid];

    const int cell = tile * CELLS_PER_TILE + tid;
    const int tix  = tIdxG[cell];
    const int ign  = ignG[cell];
    const float m  = (tix >= 0) ? 1.0f : 0.0f;
    const float ng = (ign == 0 && tix < 0) ? 1.0f : 0.0f;  // neg = conf-mask

    float tx = 0.f, ty = 0.f, tw = 0.f, th = 0.f;
    int   lbl = 0;
    if (tix >= 0) {                         // recompute target regression vals
        const float* tg = tgt + ((size_t)b * T + tix) * 5;
        float gx = tg[1] * (float)GW, gy = tg[2] * (float)GH;
        float gw = tg[3] * (float)GW, gh = tg[4] * (float)GH;
        tx = gx - floorf(gx);
        ty = gy - floorf(gy);
        tw = __logf(gw / c_AW[a] + 1e-16f);
        th = __logf(gh / c_AH[a] + 1e-16f);
        lbl = (int)tg[0];
        lbl = min(max(lbl, 0), NC - 1);
    }

    float px = sigm(p[0]), py = sigm(p[1]);
    float pw = p[2],       ph = p[3];
    float pc = sigm(p[6]);

    float s[NC], mx = -1e30f;
#pragma unroll
    for (int c = 0; c < NC; ++c) { s[c] = sigm(p[7 + c]); mx = fmaxf(mx, s[c]); }
    float se = 0.0f;
#pragma unroll
    for (int c = 0; c < NC; ++c) se += __expf(s[c] - mx);
    float lse = mx + __logf(se);
    float ce  = lse - s[lbl];               // -log_softmax(pcls)[lbl]

    float nll = -(m * logc(pc) + (1.0f - m) * logc(1.0f - pc));

    float r[NQ];
    r[0] = m;                               // n_pos
    r[1] = m * (px - tx) * (px - tx);
    r[2] = m * (py - ty) * (py - ty);
    r[3] = m * (pw - tw) * (pw - tw);
    r[4] = m * (ph - th) * (ph - th);
    r[5] = ng;                              // neg weight sum
    r[6] = ng * nll;                        // neg BCE
    r[7] = m * nll;                         // pos BCE
    r[8] = m * ce;                          // cls CE

    // ---- wave32 butterfly reduce, partials -> LDS 16x16 tile ----
    const int lane = tid & 31, wv = tid >> 5;
#pragma unroll
    for (int q = 0; q < NQ; ++q) {
        float v = r[q];
#pragma unroll
        for (int o = 16; o > 0; o >>= 1) v += __shfl_xor(v, o, 32);
        if (lane == 0) part[wv * 16 + q] = v;
    }
    __syncthreads();

    // ---- WMMA ones-GEMM reduction: D = 1(16x4) x part(4x16) + C, chained ----
    if (tid < 32) {                          // wave 0, EXEC all ones
        const int n   = tid & 15;            // column = quantity
        const int khi = tid >> 4;            // K half select
        v2f ones; ones.x = 1.0f; ones.y = 1.0f;
        v8f dacc = {};
#pragma unroll
        for (int chunk = 0; chunk < 4; ++chunk) {
            v2f bm;
            bm.x = part[(chunk * 4 + khi * 2 + 0) * 16 + n];
            bm.y = part[(chunk * 4 + khi * 2 + 1) * 16 + n];
            dacc = __builtin_amdgcn_wmma_f32_16x16x4_f32(
                false, ones, false, bm, (short)0, dacc, false, false);
        }
        // D[0][n] lives in VGPR0 of lanes 0..15 -> lane q holds total of q
        if (tid < NQ) atomicAdd(&accW[tid], dacc[0]);
    }
}

// -------------------------------------------------------- K4: finalize ------
__global__ void rl_final(const float* __restrict__ accW,
                         float* __restrict__ out, float invB) {
    float np  = fmaxf(accW[0], 1.0f);
    float mse = (accW[1] + accW[2] + accW[3] + accW[4]) / np;
    float lcf = accW[6] / fmaxf(accW[5], 1.0f) + accW[7] / fmaxf(accW[0], 1.0f);
    float lcl = invB * accW[8] / np;
    out[0] = mse + lcf + lcl;
}

// ---------------------------------------------------------------------------
extern "C" void kernel_launch(void* const* d_in, const int* in_sizes, int n_in,
                              void* d_out, int out_size, void* d_ws, size_t ws_size,
                              hipStream_t stream) {
    const float* x   = (const float*)d_in[0];
    const float* tgt = (const float*)d_in[1];

    const int B = in_sizes[0] / (NA * CH * GH * GW);   // 512
    const int T = in_sizes[1] / (B * 5);               // 50
    const int ncells = B * NA * GH * GW;               // 1,310,720

    float* accW  = (float*)d_ws;                       // 16 floats
    int*   tIdxG = (int*)((char*)d_ws + 64);
    int*   ignG  = tIdxG + ncells;

    rl_init<<<(ncells + 255) / 256, 256, 0, stream>>>(tIdxG, ignG, accW, ncells);
    rl_build_targets<<<(B * T + 255) / 256, 256, 0, stream>>>(tgt, tIdxG, ignG, B, T);
    rl_loss<<<B * NA, 512, 0, stream>>>(x, tgt, tIdxG, ignG, accW, T);
    rl_final<<<1, 1, 0, stream>>>(accW, (float*)d_out, 1.0f / (float)B);
}